// CrossAttentionMatcher_65197603553582
// MI455X (gfx1250) — compile-verified
//
#include <hip/hip_runtime.h>
#include <cmath>

#define HDIM 256
#define NHEADS 8
#define DHEAD 32
#define BB 2
#define LL1 512
#define LL2 512

typedef float v2f __attribute__((ext_vector_type(2)));
typedef float v8f __attribute__((ext_vector_type(8)));

// -------------------------------------------------------------------------
// Generic projection GEMM: out[m][n] = sum_k A[m][k] * W[n*ldw + k] + bias[n]
// M x 256 x 256, f32 WMMA 16x16x4. One wave per 16x16 output tile.
// Two accumulators (even/odd K-chunks) break the WMMA->WMMA RAW chain.
// -------------------------------------------------------------------------
__global__ void proj_gemm(const float* __restrict__ A, const float* __restrict__ W,
                          const float* __restrict__ bias, float* __restrict__ out,
                          int M, int ldw) {
  const int wid  = threadIdx.x >> 5;
  const int lane = threadIdx.x & 31;
  const int tile = blockIdx.x * 8 + wid;          // wave-uniform
  const int ntiles = (M / 16) * (HDIM / 16);
  if (tile >= ntiles) return;                     // uniform per wave -> EXEC stays all-ones

  const int ti = tile / (HDIM / 16);
  const int tn = tile % (HDIM / 16);
  const int m0 = ti * 16, n0 = tn * 16;
  const int mrow  = m0 + (lane & 15);
  const int nrow  = n0 + (lane & 15);
  const int khalf = (lane >> 4) * 2;              // 0 or 2

  const float* arow = A + (size_t)mrow * HDIM;
  const float* wrow = W + (size_t)nrow * ldw;

  v8f acc0 = {}, acc1 = {};
  #pragma unroll 4
  for (int kc = 0; kc < HDIM; kc += 8) {
    v2f a0 = *reinterpret_cast<const v2f*>(arow + kc + khalf);
    v2f b0 = *reinterpret_cast<const v2f*>(wrow + kc + khalf);
    acc0 = __builtin_amdgcn_wmma_f32_16x16x4_f32(false, a0, false, b0,
                                                 (short)0, acc0, false, false);
    v2f a1 = *reinterpret_cast<const v2f*>(arow + kc + 4 + khalf);
    v2f b1 = *reinterpret_cast<const v2f*>(wrow + kc + 4 + khalf);
    acc1 = __builtin_amdgcn_wmma_f32_16x16x4_f32(false, a1, false, b1,
                                                 (short)0, acc1, false, false);
  }
  v8f acc = acc0 + acc1;

  const float bv = bias ? bias[nrow] : 0.0f;
  const int mbase = m0 + (lane >> 4) * 8;
  const int nn = n0 + (lane & 15);
  #pragma unroll
  for (int r = 0; r < 8; ++r)
    out[(size_t)(mbase + r) * HDIM + nn] = acc[r] + bv;
}

// -------------------------------------------------------------------------
// Fused attention: logits (WMMA) -> softmax -> head-mean, all inside one
// workgroup using 256KB of CDNA5's 320KB WGP LDS. Block = 8 waves = 8 heads,
// covering 16 query rows x 512 keys. Never materializes logits in HBM.
// -------------------------------------------------------------------------
__global__ void attn_kernel(const float* __restrict__ q, const float* __restrict__ k,
                            float* __restrict__ attn) {
  extern __shared__ float sm[];                    // [8 heads][16 rows][512]
  const int b  = blockIdx.x / (LL1 / 16);
  const int i0 = (blockIdx.x % (LL1 / 16)) * 16;
  const int w    = threadIdx.x >> 5;               // head = wave id
  const int lane = threadIdx.x & 31;
  const int khalf = (lane >> 4) * 2;
  const int mloc  = lane & 15;

  // Preload q tile (16 rows x 32 dims of this head) as 8 A-fragments.
  v2f afrag[8];
  const float* qrow = q + (size_t)((b * LL1) + i0 + mloc) * HDIM + w * DHEAD;
  #pragma unroll
  for (int kc = 0; kc < 8; ++kc)
    afrag[kc] = *reinterpret_cast<const v2f*>(qrow + kc * 4 + khalf);

  const float scale = 0.17677669529663687f;        // 1/sqrt(32)
  float* myrows = sm + (size_t)w * 16 * LL2;

  for (int j0 = 0; j0 < LL2; j0 += 16) {
    const float* krow = k + (size_t)((b * LL2) + j0 + mloc) * HDIM + w * DHEAD;
    v8f acc0 = {}, acc1 = {};
    #pragma unroll
    for (int kc = 0; kc < 8; kc += 2) {
      v2f b0 = *reinterpret_cast<const v2f*>(krow + kc * 4 + khalf);
      acc0 = __builtin_amdgcn_wmma_f32_16x16x4_f32(false, afrag[kc], false, b0,
                                                   (short)0, acc0, false, false);
      v2f b1 = *reinterpret_cast<const v2f*>(krow + (kc + 1) * 4 + khalf);
      acc1 = __builtin_amdgcn_wmma_f32_16x16x4_f32(false, afrag[kc + 1], false, b1,
                                                   (short)0, acc1, false, false);
    }
    v8f acc = acc0 + acc1;
    const int rbase = (lane >> 4) * 8;
    #pragma unroll
    for (int r = 0; r < 8; ++r)
      myrows[(rbase + r) * LL2 + j0 + mloc] = acc[r] * scale;
  }
  __syncthreads();

  // Per-head softmax over 512 keys; wave32 shuffle reductions (warpSize==32).
  for (int r = 0; r < 16; ++r) {
    float* row = myrows + r * LL2;
    float mx = -3.402823466e38f;
    #pragma unroll
    for (int j = lane; j < LL2; j += 32) mx = fmaxf(mx, row[j]);
    #pragma unroll
    for (int off = 16; off; off >>= 1) mx = fmaxf(mx, __shfl_xor(mx, off));
    float s = 0.0f;
    #pragma unroll
    for (int j = lane; j < LL2; j += 32) {
      float e = __expf(row[j] - mx);
      row[j] = e;
      s += e;
    }
    #pragma unroll
    for (int off = 16; off; off >>= 1) s += __shfl_xor(s, off);
    const float inv = 1.0f / s;
    #pragma unroll
    for (int j = lane; j < LL2; j += 32) row[j] *= inv;
  }
  __syncthreads();

  // Mean over 8 heads -> attn_weights (B,L1,L2)
  for (int idx = threadIdx.x; idx < 16 * LL2; idx += 256) {
    const int r = idx >> 9;
    const int j = idx & (LL2 - 1);
    float s = 0.0f;
    #pragma unroll
    for (int h = 0; h < NHEADS; ++h) s += sm[(h * 16 + r) * LL2 + j];
    attn[(size_t)((b * LL1) + i0 + r) * LL2 + j] = s * 0.125f;
  }
}

// -------------------------------------------------------------------------
// Pairwise MLP scorer: sigmoid( sum_h relu(a1[i][h] + h2[j][h]) * w2[h] + b2 )
// ReLU inside the reduction => VALU, LDS-tiled 16x16 pairs per block.
// Tiles staged with CDNA5 async memory->LDS copies (bypass VGPRs, ASYNCcnt).
// Pitch 268 keeps the 16 distinct j-lanes on 16 distinct LDS banks and keeps
// rows 16B-aligned (268*4 = 67*16) for the B128 async writes.
// -------------------------------------------------------------------------
#define TPITCH 268
__global__ void match_kernel(const float* __restrict__ a1, const float* __restrict__ h2,
                             const float* __restrict__ W2, const float* __restrict__ b2,
                             float* __restrict__ outp) {
  __shared__ float s1[16 * TPITCH];
  __shared__ float s2[16 * TPITCH];
  __shared__ float sw[HDIM];

  const int bid = blockIdx.x;                     // B * 32 * 32 blocks
  const int b  = bid >> 10;
  const int i0 = ((bid >> 5) & 31) * 16;
  const int j0 = (bid & 31) * 16;
  const int t = threadIdx.x;

  // 1024 x 16B async copies, 4 per thread, no divergence (EXEC all-ones).
  for (int idx = t; idx < 16 * (HDIM / 4); idx += 256) {
    const int r = idx / (HDIM / 4);
    const int c = (idx % (HDIM / 4)) * 4;
    const float* g1 = a1 + (size_t)((b * LL1) + i0 + r) * HDIM + c;
    unsigned l1 = (unsigned)(size_t)&s1[r * TPITCH + c];   // low 32 bits = LDS offset
    asm volatile("global_load_async_to_lds_b128 %0, %1, off"
                 :: "v"(l1), "v"(g1) : "memory");
    const float* g2 = h2 + (size_t)((b * LL2) + j0 + r) * HDIM + c;
    unsigned l2 = (unsigned)(size_t)&s2[r * TPITCH + c];
    asm volatile("global_load_async_to_lds_b128 %0, %1, off"
                 :: "v"(l2), "v"(g2) : "memory");
  }
  if (t < HDIM / 4)
    *reinterpret_cast<float4*>(&sw[t * 4]) = *reinterpret_cast<const float4*>(W2 + t * 4);
  asm volatile("s_wait_asynccnt 0x0" ::: "memory");
  __syncthreads();

  const int li = t >> 4, lj = t & 15;
  const float* r1 = &s1[li * TPITCH];
  const float* r2 = &s2[lj * TPITCH];
  float acc = 0.0f;
  #pragma unroll 8
  for (int h = 0; h < HDIM; ++h)
    acc = fmaf(fmaxf(r1[h] + r2[h], 0.0f), sw[h], acc);

  const float z = acc + b2[0];
  outp[(size_t)((b * LL1) + i0 + li) * LL2 + j0 + lj] = 1.0f / (1.0f + __expf(-z));
}

// -------------------------------------------------------------------------
extern "C" void kernel_launch(void* const* d_in, const int* in_sizes, int n_in,
                              void* d_out, int out_size, void* d_ws, size_t ws_size,
                              hipStream_t stream) {
  const float* embed1    = (const float*)d_in[0];  // (B,L1,H)
  const float* embed2    = (const float*)d_in[1];  // (B,L2,H)
  const float* in_proj_w = (const float*)d_in[2];  // (3H,H)
  const float* in_proj_b = (const float*)d_in[3];  // (3H)
  const float* W1        = (const float*)d_in[4];  // (H,2H)
  const float* b1        = (const float*)d_in[5];  // (H)
  const float* W2        = (const float*)d_in[6];  // (1,H)
  const float* b2        = (const float*)d_in[7];  // (1)
  float* outp = (float*)d_out;                     // attn (B,L1,L2) ++ scores (B,L1,L2)

  float* ws = (float*)d_ws;                        // 4 MB of f32 scratch
  float* q  = ws;                                  // (B*L1, H)
  float* k  = q  + BB * LL1 * HDIM;                // (B*L2, H)
  float* a1 = k  + BB * LL2 * HDIM;                // h1 + b1
  float* h2 = a1 + BB * LL1 * HDIM;                // e2 @ W1[:,H:]^T

  const dim3 blk(256);
  const int M = BB * LL1;                          // 1024 rows
  const int gemmBlocks = (M / 16) * (HDIM / 16) / 8;  // 128 blocks * 8 waves

  proj_gemm<<<gemmBlocks, blk, 0, stream>>>(embed1, in_proj_w,              in_proj_b,        q,  M, HDIM);
  proj_gemm<<<gemmBlocks, blk, 0, stream>>>(embed2, in_proj_w + HDIM*HDIM,  in_proj_b + HDIM, k,  M, HDIM);
  proj_gemm<<<gemmBlocks, blk, 0, stream>>>(embed1, W1,                     b1,               a1, M, 2*HDIM);
  proj_gemm<<<gemmBlocks, blk, 0, stream>>>(embed2, W1 + HDIM,              nullptr,          h2, M, 2*HDIM);

  const size_t attnLds = (size_t)NHEADS * 16 * LL2 * sizeof(float);  // 256 KB of 320 KB WGP LDS
  attn_kernel<<<BB * (LL1 / 16), blk, attnLds, stream>>>(q, k, outp);

  match_kernel<<<BB * 32 * 32, blk, 0, stream>>>(a1, h2, W2, b2, outp + (size_t)BB * LL1 * LL2);
}